// SynthIDProcessor_22634477650270
// MI455X (gfx1250) — compile-verified
//
#include <hip/hip_runtime.h>
#include <stdint.h>

// ---------------------------------------------------------------------------
// SynthID logits processor for MI455X (gfx1250, wave32).
// One persistent 512-thread block per batch row. All block-wide float sums go
// through the matrix pipe (V_WMMA_F32_16X16X4_F32 with an all-ones B), logits
// are staged through LDS with global_load_async_to_lds_b32 / s_wait_asynccnt,
// and the top-p histogram uses LDS float atomics (ds_add_f32).
// ---------------------------------------------------------------------------

#define VOCAB   128000
#define BATCH   32
#define SEQLEN  2048
#define PRIORT  4
#define DEPTH   16
#define TOPP    0.9f
#define SEEDC   0u
#define BLOCK   512
#define NWAVES  (BLOCK / 32)
#define NBINS   2048
#define NEG_INF (-3.402823466e38f)

#define USE_ASYNC_LDS 1

#if defined(__has_builtin)
#if __has_builtin(__builtin_amdgcn_wmma_f32_16x16x4_f32)
#define HAVE_WMMA_F32X4 1
#endif
#endif

typedef float v2f __attribute__((ext_vector_type(2)));
typedef float v8f __attribute__((ext_vector_type(8)));

// ---------------- threefry2x32 (20 rounds, real rotation schedule) ---------
__device__ __forceinline__ void tf_round(uint32_t& x0, uint32_t& x1, int r) {
  x0 += x1;
  x1 = (x1 << r) | (x1 >> (32 - r));
  x1 ^= x0;
}

__device__ __forceinline__ uint2 threefry2x32(uint2 k, uint2 c) {
  uint32_t ks0 = k.x, ks1 = k.y, ks2 = k.x ^ k.y ^ 0x1BD11BDAu;
  uint32_t x0 = c.x + ks0, x1 = c.y + ks1;
  tf_round(x0, x1, 13); tf_round(x0, x1, 15); tf_round(x0, x1, 26); tf_round(x0, x1, 6);
  x0 += ks1; x1 += ks2 + 1u;
  tf_round(x0, x1, 17); tf_round(x0, x1, 29); tf_round(x0, x1, 16); tf_round(x0, x1, 24);
  x0 += ks2; x1 += ks0 + 2u;
  tf_round(x0, x1, 13); tf_round(x0, x1, 15); tf_round(x0, x1, 26); tf_round(x0, x1, 6);
  x0 += ks0; x1 += ks1 + 3u;
  tf_round(x0, x1, 17); tf_round(x0, x1, 29); tf_round(x0, x1, 16); tf_round(x0, x1, 24);
  x0 += ks1; x1 += ks2 + 4u;
  tf_round(x0, x1, 13); tf_round(x0, x1, 15); tf_round(x0, x1, 26); tf_round(x0, x1, 6);
  x0 += ks2; x1 += ks0 + 5u;
  return make_uint2(x0, x1);
}

__device__ __forceinline__ float gbit(uint2 key, uint32_t v) {
  // Bernoulli(0.5): top bit of a threefry draw.
  return (float)(threefry2x32(key, make_uint2(v, 1u)).x >> 31);
}

// ---------------- shared state --------------------------------------------
struct Smem {
  float red[BLOCK];     // reduction scratch
  float stage[BLOCK];   // async-to-LDS staging tile
  float hist[NBINS];    // top-p probability-mass histogram
  float sval[BLOCK];    // argmax values
  int   sidx[BLOCK];    // argmax indices
  uint2 dkey[DEPTH];    // per-depth g keys
  uint2 skey;           // sampling key
  float scalar;         // broadcast slot
  float tau;            // top-p threshold
  int   token;          // sampled token
};

// Block-wide f32 sum routed through the WMMA pipe:
// 512 partials -> eight 16x4 f32 A-fragments x all-ones B, chained C=D
// accumulation, then one cross-half shfl fold.
__device__ __forceinline__ float block_reduce_sum_wmma(float val, Smem& sm) {
  sm.red[threadIdx.x] = val;
  __syncthreads();
  if (threadIdx.x < 32) {               // wave 0 only: EXEC all ones
    const int lane = threadIdx.x;
    const int m    = lane & 15;
    const int kb   = (lane >> 4) << 1;  // 0 for lanes 0-15, 2 for lanes 16-31
#if defined(HAVE_WMMA_F32X4)
    v8f acc = {};
    v2f ones; ones[0] = 1.0f; ones[1] = 1.0f;
#pragma unroll
    for (int c = 0; c < BLOCK / 64; ++c) {
      v2f a;
      a[0] = sm.red[c * 64 + m * 4 + kb];
      a[1] = sm.red[c * 64 + m * 4 + kb + 1];
      // D[m][n] += A[m][0..3] * 1  (16x16x4, f32 end-to-end)
      acc = __builtin_amdgcn_wmma_f32_16x16x4_f32(
          false, a, false, ones, (short)0, acc, false, false);
    }
    float s = acc[0] + acc[1] + acc[2] + acc[3] +
              acc[4] + acc[5] + acc[6] + acc[7];   // rows 0-7 / 8-15 per half
    s += __shfl_xor(s, 16, 32);                     // fold the two halves
#else
    float s = 0.0f;
#pragma unroll 4
    for (int c = 0; c < BLOCK / 64; ++c)
      s += sm.red[c * 64 + m * 4 + kb] + sm.red[c * 64 + m * 4 + kb + 1];
    s += __shfl_xor(s, 16, 32);
#pragma unroll
    for (int o = 8; o; o >>= 1) s += __shfl_xor(s, o, 32);
#endif
    if (lane == 0) sm.scalar = s;
  }
  __syncthreads();
  float r = sm.scalar;
  __syncthreads();
  return r;
}

__device__ __forceinline__ float block_reduce_max(float val, Smem& sm) {
#pragma unroll
  for (int o = 16; o; o >>= 1) val = fmaxf(val, __shfl_xor(val, o, 32));
  if ((threadIdx.x & 31) == 0) sm.red[threadIdx.x >> 5] = val;
  __syncthreads();
  if (threadIdx.x == 0) {
    float m = sm.red[0];
    for (int w = 1; w < NWAVES; ++w) m = fmaxf(m, sm.red[w]);
    sm.scalar = m;
  }
  __syncthreads();
  float r = sm.scalar;
  __syncthreads();
  return r;
}

// ---------------------------------------------------------------------------
__global__ void __launch_bounds__(BLOCK)
synthid_row_kernel(const int* __restrict__ input_ids,
                   const float* __restrict__ logits,
                   float* __restrict__ out) {
  __shared__ Smem sm;
  const int b   = blockIdx.x;
  const int tid = threadIdx.x;
  const float* lrow = logits + (size_t)b * VOCAB;
  float*       prow = out    + (size_t)b * VOCAB;   // scratch, then scores

  // ---- key schedule (threefry fold-in chain) ----
  if (tid == 0) {
    int prior = 0;
    const int* idr = input_ids + (size_t)b * SEQLEN + (SEQLEN - PRIORT);
    for (int t = 0; t < PRIORT; ++t) prior += idr[t];
    uint2 base = make_uint2(0u, SEEDC);
    uint2 kb   = threefry2x32(base, make_uint2((uint32_t)prior, 0u));
    for (int i = 0; i < DEPTH; ++i)
      sm.dkey[i] = threefry2x32(kb, make_uint2((uint32_t)i, 0u));
    uint2 base2 = make_uint2(0u, SEEDC + 1u);
    sm.skey = threefry2x32(base2, make_uint2((uint32_t)b, 0u));
  }
  __syncthreads();

  // ---- pass A: row max, logits staged via async global->LDS copy ----
  float mx = NEG_INF;
  for (int c = 0; c < VOCAB / BLOCK; ++c) {
#if USE_ASYNC_LDS
    const float* src = lrow + c * BLOCK + tid;
    uint32_t ldsoff  = (uint32_t)(uintptr_t)&sm.stage[tid];
    asm volatile("global_load_async_to_lds_b32 %0, %1, off"
                 :: "v"(ldsoff), "v"((uint64_t)(uintptr_t)src) : "memory");
    asm volatile("s_wait_asynccnt 0" ::: "memory");
    mx = fmaxf(mx, sm.stage[tid]);   // each lane consumes its own slot
#else
    mx = fmaxf(mx, lrow[c * BLOCK + tid]);
#endif
  }
  mx = block_reduce_max(mx, sm);

  // ---- pass B: exp and softmax denominator (WMMA reduction) ----
  float psum = 0.0f;
  for (int v = tid; v < VOCAB; v += BLOCK) {
    float e = __expf(lrow[v] - mx);
    prow[v] = e;
    psum += e;
  }
  const float invS = 1.0f / block_reduce_sum_wmma(psum, sm);

  // ---- 16 sequential watermark depths: m_i = sum(g_i * p); p *= (1+g_i-m_i)
  // Each pass applies the previous depth's factor, stores p, and accumulates
  // this depth's g-mass (reduced through the matrix pipe).
  float m_prev = 0.0f;
  for (int i = 0; i < DEPTH; ++i) {
    const uint2 ki  = sm.dkey[i];
    const uint2 kp  = sm.dkey[(i > 0) ? (i - 1) : 0];
    float acc = 0.0f;
    for (int v = tid; v < VOCAB; v += BLOCK) {
      float p = prow[v];
      float f;
      if (i == 0) {
        f = invS;                                   // finish the softmax
      } else {
        f = 1.0f + gbit(kp, (uint32_t)v) - m_prev;  // previous depth's update
      }
      p *= f;
      prow[v] = p;
      acc += gbit(ki, (uint32_t)v) * p;
    }
    m_prev = block_reduce_sum_wmma(acc, sm);        // m_i
  }

  // ---- top-p: histogram of probability mass over float-ordered bins ----
  for (int j = tid; j < NBINS; j += BLOCK) sm.hist[j] = 0.0f;
  __syncthreads();
  const uint2 klast = sm.dkey[DEPTH - 1];
  float tsum = 0.0f;
  for (int v = tid; v < VOCAB; v += BLOCK) {
    float p = prow[v] * (1.0f + gbit(klast, (uint32_t)v) - m_prev);
    p = fmaxf(p, 0.0f);
    prow[v] = p;                                    // final probabilities
    tsum += p;
    uint32_t bin = __float_as_uint(p) >> 20;        // monotone for p >= 0
    if (bin > NBINS - 1) bin = NBINS - 1;
    atomicAdd(&sm.hist[bin], p);                    // ds_add_f32
  }
  const float total = block_reduce_sum_wmma(tsum, sm);
  if (tid == 0) {
    const float need = TOPP * total;
    float accum = 0.0f;
    int bin = NBINS - 1;
    while (bin > 0) {                               // sweep from high mass down
      accum += sm.hist[bin];
      if (accum >= need) break;
      --bin;
    }
    sm.tau = __uint_as_float((uint32_t)bin << 20);  // lower edge of cutoff bin
  }
  __syncthreads();
  const float tau = sm.tau;

  // ---- Gumbel-max categorical sample over the kept set {p >= tau} ----
  const uint2 sk = sm.skey;
  float best = NEG_INF;
  int   besti = 0;
  for (int v = tid; v < VOCAB; v += BLOCK) {
    float p = prow[v];
    if (p >= tau && p > 0.0f) {
      uint2 r = threefry2x32(sk, make_uint2((uint32_t)v, 0x9E3779B9u));
      float u   = ((float)(r.x >> 8) + 0.5f) * (1.0f / 16777216.0f);
      float gum = -__logf(-__logf(u));
      float sc  = __logf(p) + gum;                  // renorm shift cancels
      if (sc > best) { best = sc; besti = v; }
    }
  }
  sm.sval[tid] = best;
  sm.sidx[tid] = besti;
  __syncthreads();
  if (tid == 0) {
    float bb = sm.sval[0];
    int   bi = sm.sidx[0];
    for (int t = 1; t < BLOCK; ++t)
      if (sm.sval[t] > bb) { bb = sm.sval[t]; bi = sm.sidx[t]; }
    sm.token = bi;
  }
  __syncthreads();
  const int tok = sm.token;

  // ---- emit scores (overwrites the scratch row) ----
  for (int v = tid; v < VOCAB; v += BLOCK)
    prow[v] = (v == tok) ? 100000.0f : 1e-5f;
}

// ---------------------------------------------------------------------------
extern "C" void kernel_launch(void* const* d_in, const int* in_sizes, int n_in,
                              void* d_out, int out_size, void* d_ws,
                              size_t ws_size, hipStream_t stream) {
  (void)in_sizes; (void)n_in; (void)out_size; (void)d_ws; (void)ws_size;
  const int*   input_ids = (const int*)d_in[0];
  const float* logits    = (const float*)d_in[1];
  float*       out       = (float*)d_out;
  synthid_row_kernel<<<BATCH, BLOCK, 0, stream>>>(input_ids, logits, out);
}